// H2GCN_31164282700071
// MI455X (gfx1250) — compile-verified
//
#include <hip/hip_runtime.h>

// H2GCN inference for MI455X (gfx1250, wave32, WMMA, LDS-staged GEMM).
//
//   x   = relu(feat @ W1 + b1)            -> h[:,   0: 64]
//   x11 = scatter(x   @ Wc1) + bc1        -> h[:,  64:128]
//   x12 = scatter(x11 @ Wc1) + bc1        -> h[:, 128:192]
//   x21 = scatter(x1  @ Wc2) + bc2        -> h[:, 192:320]   (x1 = h[:,64:192])
//   x22 = scatter(x21 @ Wc2) + bc2        -> h[:, 320:448]
//   out = sigmoid(h @ W2 + b2)
//
// GEMM: V_WMMA_F32_16X16X4_F32, A/B staged through LDS (ds_load_b64 fragments,
// bank-conflict-free padding, zero-padded tails -> guard-free inner loop).
// Scatter: native global_atomic_add_f32, non-temporal gathers.

#define NN 50000
#define NE 800000
#define KB 64            // K-chunk staged per LDS round

typedef __attribute__((ext_vector_type(2))) float v2f;
typedef __attribute__((ext_vector_type(4))) float v4f;
typedef __attribute__((ext_vector_type(8))) float v8f;

// ---------------------------------------------------------------------------
// C[M x Nc] = act(A[M x K] @ B[K x Nc] + bias)
// Block = 4 waves = 64 rows; wave w owns rows [blk*64 + w*16, +16), all NT
// 16-col tiles. ACT: 0 none, 1 relu, 2 sigmoid.
//
// WMMA f32 16x16x4 fragments (wave32):
//   A: lane=(half<<4)|m  -> A[m][k + 2*half + {0,1}]       (v2f)
//   B: lane=(half<<4)|n  -> B[k + 2*half + {0,1}][n]       (v2f)
//   C: acc[j][v]         -> C[m0 + 8*half + v][j*16 + n]   (v8f)
// ---------------------------------------------------------------------------
template <int NT, int ACT>
__global__ __launch_bounds__(128) void gemm_wmma_f32(
    const float* __restrict__ A, int lda,
    const float* __restrict__ B, int ldb,
    float* __restrict__ C, int ldc,
    const float* __restrict__ bias,
    int M, int K, int Nc)
{
    constexpr int NTILE = NT * 16;
    constexpr int ASTR  = KB + 4;          // A row stride (floats): lane m -> banks 4m..4m+3
    constexpr int BSTR  = NTILE * 2 + 32;  // B pair-row stride: half-waves on disjoint banks

    __shared__ float sA[64 * ASTR];
    __shared__ float sB[(KB / 2) * BSTR];

    const int tid   = threadIdx.x;
    const int wave  = tid >> 5;
    const int lane  = tid & 31;
    const int laneM = lane & 15;
    const int half  = lane >> 4;
    const int koff  = half << 1;
    const long long mblk = (long long)blockIdx.x * 64;

    v8f acc[NT] = {};

    for (int kb = 0; kb < K; kb += KB) {
        // ---- stage A: 64 rows x KB cols, float4 coalesced, zero-padded ----
        for (int q = tid; q < 64 * (KB / 4); q += 128) {
            const int r  = q >> 4;               // KB/4 == 16 quads per row
            const int c4 = (q & 15) << 2;
            const long long row = mblk + r;
            v4f v = {0.f, 0.f, 0.f, 0.f};
            if (row < M) {
                const int kg = kb + c4;
                if (kg + 3 < K) {
                    v = *(const v4f*)(A + row * (long long)lda + kg);
                } else {
                    for (int u = 0; u < 4; ++u)
                        if (kg + u < K) v[u] = A[row * (long long)lda + kg + u];
                }
            }
            float* d = sA + r * ASTR + c4;
            d[0] = v.x; d[1] = v.y; d[2] = v.z; d[3] = v.w;
        }
        // ---- stage B: KB rows x NTILE cols, pair-interleaved, zero-padded ----
        for (int q = tid; q < KB * (NTILE / 2); q += 128) {
            const int r  = q / (NTILE / 2);
            const int c  = (q - r * (NTILE / 2)) * 2;
            const int kg = kb + r;
            float b0 = 0.f, b1 = 0.f;
            if (kg < K) {
                const float* bp = B + (long long)kg * ldb + c;
                if (c     < Nc) b0 = bp[0];
                if (c + 1 < Nc) b1 = bp[1];
            }
            float* d = sB + (r >> 1) * BSTR + c * 2 + (r & 1);
            d[0] = b0;
            d[2] = b1;
        }
        __syncthreads();

        // ---- guard-free WMMA loop: 1 + NT ds_load_b64, NT wmma per 4 K ----
        const float* aw = sA + (wave * 16 + laneM) * ASTR + koff;
#pragma unroll 4
        for (int kp = 0; kp < KB; kp += 4) {
            const v2f a = *(const v2f*)(aw + kp);
            const float* bp = sB + ((kp >> 1) + half) * BSTR + laneM * 2;
#pragma unroll
            for (int j = 0; j < NT; ++j) {
                const v2f b = *(const v2f*)(bp + j * 32);
                acc[j] = __builtin_amdgcn_wmma_f32_16x16x4_f32(
                    false, a, false, b, (short)0, acc[j], false, false);
            }
        }
        __syncthreads();
    }

    // ---- epilogue: bias + activation, tail-guarded stores ----
    const long long rb = mblk + wave * 16 + half * 8;
#pragma unroll
    for (int j = 0; j < NT; ++j) {
        const int n = j * 16 + laneM;
        if (n < Nc) {
            const float bv = bias ? bias[n] : 0.0f;
#pragma unroll
            for (int v = 0; v < 8; ++v) {
                const long long r = rb + v;
                if (r < M) {
                    float x = acc[j][v] + bv;
                    if (ACT == 1) x = fmaxf(x, 0.0f);
                    if (ACT == 2) x = 1.0f / (1.0f + __expf(-x));
                    C[r * (long long)ldc + n] = x;
                }
            }
        }
    }
}

// ---------------------------------------------------------------------------
// agg[i, 0:D) = b[0:D)   (bias pre-load before atomic scatter)
// ---------------------------------------------------------------------------
template <int D>
__global__ __launch_bounds__(256) void init_bias(
    float* __restrict__ out, int ldo, const float* __restrict__ b)
{
    long long idx = (long long)blockIdx.x * blockDim.x + threadIdx.x;
    long long i = idx / D;
    int c = (int)(idx - i * D);
    if (i >= NN) return;
    out[i * ldo + c] = b[c];
}

// ---------------------------------------------------------------------------
// agg[dst[e], c] += t[src[e], c]  via native fp32 atomics, float4 gathers.
// D4 = d/4 column-quads; 16 (d=64) or 32 (d=128) lanes cooperate per edge.
// ---------------------------------------------------------------------------
template <int D4>
__global__ __launch_bounds__(256) void scatter_add(
    const float* __restrict__ t, int ldt,
    const long long* __restrict__ ei,   // [2, NE] int64: row0=src, row1=dst
    float* __restrict__ agg, int ldagg)
{
    long long idx = (long long)blockIdx.x * blockDim.x + threadIdx.x;
    long long e = idx / D4;
    int c = (int)(idx - e * D4) * 4;
    if (e >= NE) return;
    long long s = ei[e];
    long long d = ei[(long long)NE + e];
    const v4f v = __builtin_nontemporal_load((const v4f*)(t + s * (long long)ldt + c));
    float* o = agg + d * (long long)ldagg + c;
    unsafeAtomicAdd(o + 0, v.x);   // -> global_atomic_add_f32
    unsafeAtomicAdd(o + 1, v.y);
    unsafeAtomicAdd(o + 2, v.z);
    unsafeAtomicAdd(o + 3, v.w);
}

// ---------------------------------------------------------------------------
extern "C" void kernel_launch(void* const* d_in, const int* in_sizes, int n_in,
                              void* d_out, int out_size, void* d_ws, size_t ws_size,
                              hipStream_t stream)
{
    const float*      feat = (const float*)d_in[0];       // [NN, 500]
    const long long*  ei   = (const long long*)d_in[1];   // [2, NE] int64
    const float*      W1   = (const float*)d_in[2];       // [500, 64]
    const float*      b1   = (const float*)d_in[3];       // [64]
    const float*      Wc1  = (const float*)d_in[4];       // [64, 64]
    const float*      bc1  = (const float*)d_in[5];       // [64]
    const float*      Wc2  = (const float*)d_in[6];       // [128, 128]
    const float*      bc2  = (const float*)d_in[7];       // [128]
    const float*      W2   = (const float*)d_in[8];       // [448, 50]
    const float*      b2   = (const float*)d_in[9];       // [50]
    float*            out  = (float*)d_out;               // [NN, 50]

    // Workspace: h [NN x 448] concat buffer, t [NN x 128] GEMM temp.
    float* h = (float*)d_ws;
    float* t = h + (size_t)NN * 448;
    const int LDH = 448;

    const dim3 gblk(128);
    const int  ggrid = (NN + 63) / 64;                         // 782 blocks
    const int  ib64  = (int)(((long long)NN * 64  + 255) / 256);
    const int  ib128 = (int)(((long long)NN * 128 + 255) / 256);
    const int  sc64  = (int)(((long long)NE * 16  + 255) / 256);
    const int  sc128 = (int)(((long long)NE * 32  + 255) / 256);

    // x = relu(feat @ W1 + b1) -> h[:, 0:64]
    gemm_wmma_f32<4, 1><<<ggrid, gblk, 0, stream>>>(feat, 500, W1, 64, h, LDH, b1, NN, 500, 64);

    // x11 = scatter(x @ Wc1) + bc1 -> h[:, 64:128]
    gemm_wmma_f32<4, 0><<<ggrid, gblk, 0, stream>>>(h, LDH, Wc1, 64, t, 64, nullptr, NN, 64, 64);
    init_bias<64><<<ib64, 256, 0, stream>>>(h + 64, LDH, bc1);
    scatter_add<16><<<sc64, 256, 0, stream>>>(t, 64, ei, h + 64, LDH);

    // x12 = scatter(x11 @ Wc1) + bc1 -> h[:, 128:192]
    gemm_wmma_f32<4, 0><<<ggrid, gblk, 0, stream>>>(h + 64, LDH, Wc1, 64, t, 64, nullptr, NN, 64, 64);
    init_bias<64><<<ib64, 256, 0, stream>>>(h + 128, LDH, bc1);
    scatter_add<16><<<sc64, 256, 0, stream>>>(t, 64, ei, h + 128, LDH);

    // x21 = scatter(x1 @ Wc2) + bc2 -> h[:, 192:320]   (x1 = h[:, 64:192], K=128)
    gemm_wmma_f32<8, 0><<<ggrid, gblk, 0, stream>>>(h + 64, LDH, Wc2, 128, t, 128, nullptr, NN, 128, 128);
    init_bias<128><<<ib128, 256, 0, stream>>>(h + 192, LDH, bc2);
    scatter_add<32><<<sc128, 256, 0, stream>>>(t, 128, ei, h + 192, LDH);

    // x22 = scatter(x21 @ Wc2) + bc2 -> h[:, 320:448]
    gemm_wmma_f32<8, 0><<<ggrid, gblk, 0, stream>>>(h + 192, LDH, Wc2, 128, t, 128, nullptr, NN, 128, 128);
    init_bias<128><<<ib128, 256, 0, stream>>>(h + 320, LDH, bc2);
    scatter_add<32><<<sc128, 256, 0, stream>>>(t, 128, ei, h + 320, LDH);

    // out = sigmoid(h @ W2 + b2), Nc=50 padded to one 64-wide tile
    gemm_wmma_f32<4, 2><<<ggrid, gblk, 0, stream>>>(h, LDH, W2, 50, out, 50, b2, NN, 448, 50);
}